// QwenMoE_75935021793920
// MI455X (gfx1250) — compile-verified
//
#include <hip/hip_runtime.h>
#include <hip/hip_bf16.h>

// ---------------- problem constants ----------------
#define T_TOK  256
#define H_DIM  2048
#define E_NUM  60
#define K_TOP  4
#define F_DIM  1408
#define FS_DIM 5632
#define NSLOT  (T_TOK * K_TOP)   // 1024, every slot always used

typedef __attribute__((ext_vector_type(2))) float v2f;
typedef __attribute__((ext_vector_type(8))) float v8f;

// D = A(16x4 f32) * B(4x16 f32) + C(16x16 f32), full fp32 precision.
__device__ __forceinline__ v8f wmma4(v2f a, v2f b, v8f c) {
  return __builtin_amdgcn_wmma_f32_16x16x4_f32(
      /*neg_a=*/false, a, /*neg_b=*/false, b,
      /*c_mod=*/(short)0, c, /*reuse_a=*/false, /*reuse_b=*/false);
}

// Single-use weight streams: non-temporal so they don't evict the reused
// buffers (x, a_buf, partial, s_act) from the 192MB L2.
__device__ __forceinline__ float ntl(const float* p) {
  return __builtin_nontemporal_load(p);
}

__device__ __forceinline__ float silu_f(float g) {
  return g / (1.f + expf(-g));
}

// ---------------- kernel 1: router (softmax + top4) + shared sigmoid gate ----
__global__ __launch_bounds__(64)
void router_kernel(const float* __restrict__ x,
                   const float* __restrict__ gate_w,
                   const float* __restrict__ shared_gate_w,
                   int* __restrict__ topk_idx,
                   float* __restrict__ topk_val,
                   float* __restrict__ sig) {
  const int t = blockIdx.x;
  const int tid = threadIdx.x;
  __shared__ float logits[E_NUM];
  __shared__ float red[64];
  const float* xr = x + (size_t)t * H_DIM;

  // shared-expert sigmoid gate: dot(x[t], shared_gate_w)
  float p = 0.f;
  for (int h = tid; h < H_DIM; h += 64) p += xr[h] * shared_gate_w[h];
  red[tid] = p;
  __syncthreads();
  if (tid == 0) {
    float s = 0.f;
    for (int i = 0; i < 64; ++i) s += red[i];
    sig[t] = 1.f / (1.f + expf(-s));
  }

  // router logits: thread e computes dot(x[t], gate_w[:, e]); coalesced in e
  if (tid < E_NUM) {
    float acc = 0.f;
    for (int h = 0; h < H_DIM; ++h) acc += xr[h] * gate_w[(size_t)h * E_NUM + tid];
    logits[tid] = acc;
  }
  __syncthreads();

  if (tid == 0) {
    float mx = logits[0];
    for (int e = 1; e < E_NUM; ++e) mx = fmaxf(mx, logits[e]);
    float pr[E_NUM];
    float sum = 0.f;
    for (int e = 0; e < E_NUM; ++e) { pr[e] = expf(logits[e] - mx); sum += pr[e]; }
    const float inv = 1.f / sum;
    for (int e = 0; e < E_NUM; ++e) pr[e] *= inv;
    for (int k = 0; k < K_TOP; ++k) {         // first-index tie-break, like lax.top_k
      int bi = 0; float bv = -1.f;
      for (int e = 0; e < E_NUM; ++e) if (pr[e] > bv) { bv = pr[e]; bi = e; }
      topk_idx[t * K_TOP + k] = bi;
      topk_val[t * K_TOP + k] = bv;
      pr[bi] = -2.f;
    }
  }
}

// ---------------- kernel 2: deterministic expert grouping ------------------
__global__ __launch_bounds__(64)
void group_kernel(const int* __restrict__ topk_idx,
                  const float* __restrict__ topk_val,
                  int* __restrict__ cnt, int* __restrict__ base,
                  int* __restrict__ tok_of_slot, float* __restrict__ wt_of_slot,
                  int* __restrict__ slot_of_tk) {
  const int e = threadIdx.x;
  __shared__ int s_base[E_NUM];
  if (e < E_NUM) {
    int c = 0;
    for (int i = 0; i < NSLOT; ++i) c += (topk_idx[i] == e);
    cnt[e] = c;
  }
  __syncthreads();
  if (e == 0) {
    int acc = 0;
    for (int i = 0; i < E_NUM; ++i) { s_base[i] = acc; base[i] = acc; acc += cnt[i]; }
  }
  __syncthreads();
  if (e < E_NUM) {
    int pos = s_base[e];
    for (int i = 0; i < NSLOT; ++i) {
      if (topk_idx[i] == e) {
        tok_of_slot[pos] = i >> 2;       // token = i / K_TOP
        wt_of_slot[pos]  = topk_val[i];
        slot_of_tk[i]    = pos;
        ++pos;
      }
    }
  }
}

// ---------------- kernel 3: expert gate/up + SiLU*up*weight ----------------
// grid (E_NUM, F_DIM/64), block 128 (4 waves, each owns a 16-col N-tile).
__global__ __launch_bounds__(128)
void expert_gateup_kernel(const float* __restrict__ x,
                          const float* __restrict__ w_gate,
                          const float* __restrict__ w_up,
                          const int* __restrict__ cnt,
                          const int* __restrict__ base,
                          const int* __restrict__ tok_of_slot,
                          const float* __restrict__ wt_of_slot,
                          float* __restrict__ a_buf) {
  const int e = blockIdx.x;
  const int Me = cnt[e];
  if (Me == 0) return;
  const int b0 = base[e];
  const int wave  = threadIdx.x >> 5;
  const int lane  = threadIdx.x & 31;
  const int laneM = lane & 15;
  const int laneK = (lane >> 4) << 1;              // 0 or 2
  const int col   = blockIdx.y * 64 + wave * 16 + laneM;
  const float* bg = w_gate + (size_t)e * H_DIM * F_DIM + col;
  const float* bu = w_up   + (size_t)e * H_DIM * F_DIM + col;

  __shared__ int   s_tok[T_TOK];
  __shared__ float s_wt[T_TOK];
  for (int i = threadIdx.x; i < Me; i += 128) {
    s_tok[i] = tok_of_slot[b0 + i];
    s_wt[i]  = wt_of_slot[b0 + i];
  }
  __syncthreads();

  for (int mt = 0; mt < Me; mt += 32) {            // 2 M-tiles per weight pass
    const int r0 = mt + laneM, r1 = mt + 16 + laneM;
    const bool v0 = r0 < Me, v1 = r1 < Me;
    const float* xr0 = x + (size_t)s_tok[v0 ? r0 : 0] * H_DIM + laneK;
    const float* xr1 = x + (size_t)s_tok[v1 ? r1 : 0] * H_DIM + laneK;
    v8f g0 = {}, g1 = {}, u0 = {}, u1 = {};
    for (int k = 0; k < H_DIM; k += 4) {
      v2f a0 = *(const v2f*)(xr0 + k);
      v2f a1 = *(const v2f*)(xr1 + k);
      if (!v0) { a0.x = 0.f; a0.y = 0.f; }
      if (!v1) { a1.x = 0.f; a1.y = 0.f; }
      const float* bgk = bg + (size_t)(k + laneK) * F_DIM;
      const float* buk = bu + (size_t)(k + laneK) * F_DIM;
      v2f vbg; vbg.x = ntl(bgk); vbg.y = ntl(bgk + F_DIM);
      v2f vbu; vbu.x = ntl(buk); vbu.y = ntl(buk + F_DIM);
      g0 = wmma4(a0, vbg, g0);
      g1 = wmma4(a1, vbg, g1);
      u0 = wmma4(a0, vbu, u0);
      u1 = wmma4(a1, vbu, u1);
    }
    const int mrow = (lane >> 4) << 3;             // 0 or 8
#pragma unroll
    for (int r = 0; r < 8; ++r) {
      const int row = mt + mrow + r;
      if (row < Me)
        a_buf[(size_t)(b0 + row) * F_DIM + col] = silu_f(g0[r]) * u0[r] * s_wt[row];
      const int row2 = row + 16;
      if (row2 < Me)
        a_buf[(size_t)(b0 + row2) * F_DIM + col] = silu_f(g1[r]) * u1[r] * s_wt[row2];
    }
  }
}

// ---------------- kernel 4: expert down-proj -> per-slot partials ----------
// grid (E_NUM, H_DIM/64), block 128.
__global__ __launch_bounds__(128)
void expert_down_kernel(const float* __restrict__ a_buf,
                        const float* __restrict__ w_down,
                        const int* __restrict__ cnt,
                        const int* __restrict__ base,
                        float* __restrict__ partial) {
  const int e = blockIdx.x;
  const int Me = cnt[e];
  if (Me == 0) return;
  const int b0 = base[e];
  const int wave  = threadIdx.x >> 5;
  const int lane  = threadIdx.x & 31;
  const int laneM = lane & 15;
  const int laneK = (lane >> 4) << 1;
  const int col   = blockIdx.y * 64 + wave * 16 + laneM;
  const float* bd = w_down + (size_t)e * F_DIM * H_DIM + col;

  for (int mt = 0; mt < Me; mt += 32) {
    const int r0 = mt + laneM, r1 = mt + 16 + laneM;
    const bool v0 = r0 < Me, v1 = r1 < Me;
    const float* ar0 = a_buf + (size_t)(b0 + (v0 ? r0 : 0)) * F_DIM + laneK;
    const float* ar1 = a_buf + (size_t)(b0 + (v1 ? r1 : 0)) * F_DIM + laneK;
    v8f d0 = {}, d1 = {};
    for (int k = 0; k < F_DIM; k += 4) {
      v2f a0 = *(const v2f*)(ar0 + k);
      v2f a1 = *(const v2f*)(ar1 + k);
      if (!v0) { a0.x = 0.f; a0.y = 0.f; }
      if (!v1) { a1.x = 0.f; a1.y = 0.f; }
      const float* bdk = bd + (size_t)(k + laneK) * H_DIM;
      v2f vb; vb.x = ntl(bdk); vb.y = ntl(bdk + H_DIM);
      d0 = wmma4(a0, vb, d0);
      d1 = wmma4(a1, vb, d1);
    }
    const int mrow = (lane >> 4) << 3;
#pragma unroll
    for (int r = 0; r < 8; ++r) {
      const int row = mt + mrow + r;
      if (row < Me)  partial[(size_t)(b0 + row)  * H_DIM + col] = d0[r];
      const int row2 = row + 16;
      if (row2 < Me) partial[(size_t)(b0 + row2) * H_DIM + col] = d1[r];
    }
  }
}

// ---------------- kernel 5: shared expert gate/up activation ---------------
// grid (FS_DIM/64, T_TOK/64), block 128; 4 M-tiles per block.
__global__ __launch_bounds__(128)
void shared_act_kernel(const float* __restrict__ x,
                       const float* __restrict__ sw_gate,
                       const float* __restrict__ sw_up,
                       float* __restrict__ s_act) {
  const int wave  = threadIdx.x >> 5;
  const int lane  = threadIdx.x & 31;
  const int laneM = lane & 15;
  const int laneK = (lane >> 4) << 1;
  const int col   = blockIdx.x * 64 + wave * 16 + laneM;
  const int mt    = blockIdx.y * 64;
  const float* bg = sw_gate + col;
  const float* bu = sw_up + col;

  v8f g[4] = {{}, {}, {}, {}};
  v8f u[4] = {{}, {}, {}, {}};
  for (int k = 0; k < H_DIM; k += 4) {
    const float* bgk = bg + (size_t)(k + laneK) * FS_DIM;
    const float* buk = bu + (size_t)(k + laneK) * FS_DIM;
    v2f vbg; vbg.x = ntl(bgk); vbg.y = ntl(bgk + FS_DIM);
    v2f vbu; vbu.x = ntl(buk); vbu.y = ntl(buk + FS_DIM);
#pragma unroll
    for (int j = 0; j < 4; ++j) {
      v2f a = *(const v2f*)(x + (size_t)(mt + j * 16 + laneM) * H_DIM + k + laneK);
      g[j] = wmma4(a, vbg, g[j]);
      u[j] = wmma4(a, vbu, u[j]);
    }
  }
  const int mrow = (lane >> 4) << 3;
#pragma unroll
  for (int j = 0; j < 4; ++j)
#pragma unroll
    for (int r = 0; r < 8; ++r) {
      const int t = mt + j * 16 + mrow + r;
      s_act[(size_t)t * FS_DIM + col] = silu_f(g[j][r]) * u[j][r];
    }
}

// ------- kernel 6: shared down-proj * sigmoid gate + expert combine --------
// grid (H_DIM/64, T_TOK/128), block 128; 8 M-tiles per block (halves the
// sw_down re-read traffic vs 4 M-tiles).
__global__ __launch_bounds__(128)
void shared_down_combine_kernel(const float* __restrict__ s_act,
                                const float* __restrict__ sw_down,
                                const float* __restrict__ partial,
                                const int* __restrict__ slot_of_tk,
                                const float* __restrict__ sig,
                                float* __restrict__ out) {
  const int wave  = threadIdx.x >> 5;
  const int lane  = threadIdx.x & 31;
  const int laneM = lane & 15;
  const int laneK = (lane >> 4) << 1;
  const int col   = blockIdx.x * 64 + wave * 16 + laneM;
  const int mt    = blockIdx.y * 128;
  const float* bd = sw_down + col;

  __shared__ int   s_slot[128 * K_TOP];
  __shared__ float s_sig[128];
  for (int i = threadIdx.x; i < 128 * K_TOP; i += 128)
    s_slot[i] = slot_of_tk[mt * K_TOP + i];
  for (int i = threadIdx.x; i < 128; i += 128)
    s_sig[i] = sig[mt + i];
  __syncthreads();

  v8f d[8] = {{}, {}, {}, {}, {}, {}, {}, {}};
  for (int k = 0; k < FS_DIM; k += 4) {
    const float* bdk = bd + (size_t)(k + laneK) * H_DIM;
    v2f vb; vb.x = ntl(bdk); vb.y = ntl(bdk + H_DIM);
#pragma unroll
    for (int j = 0; j < 8; ++j) {
      v2f a = *(const v2f*)(s_act + (size_t)(mt + j * 16 + laneM) * FS_DIM + k + laneK);
      d[j] = wmma4(a, vb, d[j]);
    }
  }
  const int mrow = (lane >> 4) << 3;
#pragma unroll
  for (int j = 0; j < 8; ++j)
#pragma unroll
    for (int r = 0; r < 8; ++r) {
      const int tl = j * 16 + mrow + r;          // token index within block chunk
      const int t  = mt + tl;
      float acc = d[j][r] * s_sig[tl];
      const int* sl = s_slot + tl * K_TOP;       // fixed-order sum -> deterministic
      acc += partial[(size_t)sl[0] * H_DIM + col];
      acc += partial[(size_t)sl[1] * H_DIM + col];
      acc += partial[(size_t)sl[2] * H_DIM + col];
      acc += partial[(size_t)sl[3] * H_DIM + col];
      out[(size_t)t * H_DIM + col] = acc;
    }
}

// ---------------- host launcher --------------------------------------------
extern "C" void kernel_launch(void* const* d_in, const int* in_sizes, int n_in,
                              void* d_out, int out_size, void* d_ws, size_t ws_size,
                              hipStream_t stream) {
  (void)in_sizes; (void)n_in; (void)out_size; (void)ws_size;
  const float* x             = (const float*)d_in[0];
  const float* gate_w        = (const float*)d_in[1];
  const float* w_gate        = (const float*)d_in[2];
  const float* w_up          = (const float*)d_in[3];
  const float* w_down        = (const float*)d_in[4];
  const float* sw_gate       = (const float*)d_in[5];
  const float* sw_up         = (const float*)d_in[6];
  const float* sw_down       = (const float*)d_in[7];
  const float* shared_gate_w = (const float*)d_in[8];
  float* out = (float*)d_out;

  char* ws = (char*)d_ws;
  size_t off = 0;
  auto carve = [&](size_t bytes) -> void* {
    void* p = ws + off;
    off = (off + bytes + 255) & ~(size_t)255;
    return p;
  };
  float* a_buf       = (float*)carve((size_t)NSLOT * F_DIM * sizeof(float));   // 5.8 MB
  float* partial     = (float*)carve((size_t)NSLOT * H_DIM * sizeof(float));   // 8.4 MB
  float* s_act       = (float*)carve((size_t)T_TOK * FS_DIM * sizeof(float));  // 5.8 MB
  float* sig         = (float*)carve(T_TOK * sizeof(float));
  float* topk_val    = (float*)carve(NSLOT * sizeof(float));
  float* wt_of_slot  = (float*)carve(NSLOT * sizeof(float));
  int*   topk_idx    = (int*)carve(NSLOT * sizeof(int));
  int*   tok_of_slot = (int*)carve(NSLOT * sizeof(int));
  int*   slot_of_tk  = (int*)carve(NSLOT * sizeof(int));
  int*   cnt         = (int*)carve(E_NUM * sizeof(int));
  int*   base        = (int*)carve(E_NUM * sizeof(int));

  router_kernel<<<T_TOK, 64, 0, stream>>>(x, gate_w, shared_gate_w,
                                          topk_idx, topk_val, sig);
  group_kernel<<<1, 64, 0, stream>>>(topk_idx, topk_val, cnt, base,
                                     tok_of_slot, wt_of_slot, slot_of_tk);
  expert_gateup_kernel<<<dim3(E_NUM, F_DIM / 64), 128, 0, stream>>>(
      x, w_gate, w_up, cnt, base, tok_of_slot, wt_of_slot, a_buf);
  expert_down_kernel<<<dim3(E_NUM, H_DIM / 64), 128, 0, stream>>>(
      a_buf, w_down, cnt, base, partial);
  shared_act_kernel<<<dim3(FS_DIM / 64, T_TOK / 64), 128, 0, stream>>>(
      x, sw_gate, sw_up, s_act);
  shared_down_combine_kernel<<<dim3(H_DIM / 64, T_TOK / 128), 128, 0, stream>>>(
      s_act, sw_down, partial, slot_of_tk, sig, out);
}